// PairwiseContactPredictionHead_36378372997753
// MI455X (gfx1250) — compile-verified
//
#include <hip/hip_runtime.h>

// out[b,i,j] = sum_d X[b,i,d]*w1[d]*X[b,j,d] + b0
// (exactly equal to the reference's symmetrized pred: the lin_i - lin_j term
//  is antisymmetric and cancels under (pred + pred^T)/2; bilin is symmetric.)

typedef __attribute__((ext_vector_type(2))) float v2f;
typedef __attribute__((ext_vector_type(8))) float v8f;

#define L_DIM 512
#define D_DIM 256
#define TILE  64     // block output tile: TILE x TILE
#define KC    64     // K chunk staged in LDS
#define LDSP  68     // padded LDS row stride (floats): 68 % 64 banks = 4 -> conflict-free frag loads

__global__ __launch_bounds__(128) void
pairwise_bilin_wmma_f32(const float* __restrict__ X,   // (2, 512, 256)
                        const float* __restrict__ W,   // (512, 1): w1 = W[0:256]
                        const float* __restrict__ bias,// (1,)
                        float* __restrict__ out)       // (2, 512, 512)
{
    __shared__ __align__(16) float As[TILE * LDSP]; // rows I0.. scaled by w1
    __shared__ __align__(16) float Bs[TILE * LDSP]; // rows J0.. raw

    const int bi = blockIdx.z;
    const int I0 = blockIdx.y * TILE;
    const int J0 = blockIdx.x * TILE;
    const float b0 = bias[0];

    const int t    = threadIdx.x;      // 0..127
    const int wave = t >> 5;           // 0..3
    const int lane = t & 31;
    const int wr   = (wave & 1) * 32;  // wave row offset inside block tile
    const int wc   = (wave >> 1) * 32; // wave col offset inside block tile
    const int m    = lane & 15;        // M (A) / N (B) index
    const int kp   = (lane >> 4) * 2;  // K pair base: 0 or 2

    const float* Xb = X + (size_t)bi * L_DIM * D_DIM;

    v8f acc[2][2] = {};

    for (int k0 = 0; k0 < D_DIM; k0 += KC) {
        // ---- cooperative panel load: 64 rows x 64 floats per panel, float4 chunks ----
        // 1024 chunks / 128 threads = 8 per thread.
        for (int c = t; c < TILE * (KC / 4); c += 128) {
            const int row  = c >> 4;          // chunk / 16
            const int col4 = (c & 15) << 2;   // 4-float column offset

            const float4 xa = *(const float4*)(Xb + (size_t)(I0 + row) * D_DIM + k0 + col4);
            const float4 xb = *(const float4*)(Xb + (size_t)(J0 + row) * D_DIM + k0 + col4);
            const float4 wv = *(const float4*)(W + k0 + col4);  // w1 slice

            float4 av;
            av.x = xa.x * wv.x; av.y = xa.y * wv.y;
            av.z = xa.z * wv.z; av.w = xa.w * wv.w;

            *(float4*)&As[row * LDSP + col4] = av;   // 16B aligned: LDSP%4==0, col4%4==0
            *(float4*)&Bs[row * LDSP + col4] = xb;
        }
        __syncthreads();

        // ---- 16 k-steps of V_WMMA_F32_16X16X4_F32, 2x2 tile block per wave ----
        #pragma unroll
        for (int kk = 0; kk < KC; kk += 4) {
            // A frag (16x4 f32): lane m holds row M=m, VGPR pair = K {kp, kp+1}
            v2f a0 = *(const v2f*)&As[(wr + m)      * LDSP + kk + kp];
            v2f a1 = *(const v2f*)&As[(wr + 16 + m) * LDSP + kk + kp];
            // B frag (4x16 f32) of X^T: B[k][n] = X[J0+..+n][k]
            v2f bq0 = *(const v2f*)&Bs[(wc + m)      * LDSP + kk + kp];
            v2f bq1 = *(const v2f*)&Bs[(wc + 16 + m) * LDSP + kk + kp];

            acc[0][0] = __builtin_amdgcn_wmma_f32_16x16x4_f32(false, a0, false, bq0,
                                                              (short)0, acc[0][0], false, false);
            acc[0][1] = __builtin_amdgcn_wmma_f32_16x16x4_f32(false, a0, false, bq1,
                                                              (short)0, acc[0][1], false, false);
            acc[1][0] = __builtin_amdgcn_wmma_f32_16x16x4_f32(false, a1, false, bq0,
                                                              (short)0, acc[1][0], false, false);
            acc[1][1] = __builtin_amdgcn_wmma_f32_16x16x4_f32(false, a1, false, bq1,
                                                              (short)0, acc[1][1], false, false);
        }
        __syncthreads();
    }

    // ---- store: C/D layout -> VGPR r holds M = r + 8*(lane/16), N = lane%16 ----
    float* outb = out + (size_t)bi * L_DIM * L_DIM;
    const int rsel = (lane >> 4) * 8;
    #pragma unroll
    for (int ti = 0; ti < 2; ti++) {
        #pragma unroll
        for (int tj = 0; tj < 2; tj++) {
            const int rowBase = I0 + wr + ti * 16 + rsel;
            const int col     = J0 + wc + tj * 16 + m;
            #pragma unroll
            for (int r = 0; r < 8; r++) {
                outb[(size_t)(rowBase + r) * L_DIM + col] = acc[ti][tj][r] + b0;
            }
        }
    }
}

extern "C" void kernel_launch(void* const* d_in, const int* in_sizes, int n_in,
                              void* d_out, int out_size, void* d_ws, size_t ws_size,
                              hipStream_t stream) {
    (void)in_sizes; (void)n_in; (void)out_size; (void)d_ws; (void)ws_size;
    const float* X    = (const float*)d_in[0]; // (2,512,256) f32
    const float* W    = (const float*)d_in[1]; // (512,1) f32
    const float* bias = (const float*)d_in[2]; // (1,) f32
    float* out = (float*)d_out;                // (2,512,512,1) f32

    dim3 grid(L_DIM / TILE, L_DIM / TILE, 2);  // (8, 8, 2)
    dim3 block(128);                            // 4 waves (wave32)
    pairwise_bilin_wmma_f32<<<grid, block, 0, stream>>>(X, W, bias, out);
}